// AdaptiveAveragePooling_11819749998719
// MI455X (gfx1250) — compile-verified
//
#include <hip/hip_runtime.h>

// Problem constants (from reference): x is (B=16, H=512, W=512, C=64) f32, out is (16,7,7,64) f32.
#define B_  16
#define H_  512
#define W_  512
#define C_  64
#define OH_ 7
#define OW_ 7

typedef __attribute__((ext_vector_type(2))) float v2f;
typedef __attribute__((ext_vector_type(8))) float v8f;

__device__ __forceinline__ int win_start(int o, int in) { return (o * in) / 7; }
__device__ __forceinline__ int win_stop (int o, int in) { return ((o + 1) * in + 6) / 7; }

// ---------------------------------------------------------------------------
// Pass 1: reduce H.  y1[b,oh,w,c] = (1/sizeH) * sum_{h in win(oh)} x[b,h,w,c]
// Grid: B*OH*32 blocks, 256 threads. Each thread owns one float4 column
// (w*16 + c4) and streams ~74 rows. Fully coalesced b128 loads.
// ---------------------------------------------------------------------------
__global__ __launch_bounds__(256) void pool_h_kernel(const float* __restrict__ x,
                                                     float* __restrict__ y1) {
    const int blk  = blockIdx.x;
    const int tile = blk & 31;        // 32 tiles of 256 float4-columns
    const int boh  = blk >> 5;        // b*7 + oh
    const int oh   = boh % OH_;
    const int b    = boh / OH_;

    const int col4 = tile * 256 + threadIdx.x;      // 0..8191 : w*16 + c4
    const int h0   = win_start(oh, H_);
    const int h1   = win_stop(oh, H_);
    const float inv = 1.0f / (float)(h1 - h0);

    const float4* __restrict__ p =
        (const float4*)x + (size_t)b * (H_ * W_ * (C_ / 4)) + (size_t)h0 * (W_ * (C_ / 4)) + col4;
    const int rowStride = W_ * (C_ / 4);            // 8192 float4 per H row

    float4 a0 = make_float4(0.f, 0.f, 0.f, 0.f);
    float4 a1 = make_float4(0.f, 0.f, 0.f, 0.f);
    int h = h0;
    for (; h + 1 < h1; h += 2) {
        float4 v0 = p[0];
        float4 v1 = p[rowStride];
        a0.x += v0.x; a0.y += v0.y; a0.z += v0.z; a0.w += v0.w;
        a1.x += v1.x; a1.y += v1.y; a1.z += v1.z; a1.w += v1.w;
        p += 2 * rowStride;
    }
    if (h < h1) {
        float4 v0 = p[0];
        a0.x += v0.x; a0.y += v0.y; a0.z += v0.z; a0.w += v0.w;
    }
    float4 s;
    s.x = (a0.x + a1.x) * inv;
    s.y = (a0.y + a1.y) * inv;
    s.z = (a0.z + a1.z) * inv;
    s.w = (a0.w + a1.w) * inv;

    float4* __restrict__ yp = (float4*)y1 + (size_t)boh * (W_ * (C_ / 4)) + col4;
    *yp = s;
}

// ---------------------------------------------------------------------------
// Pass 2: reduce W via V_WMMA_F32_16X16X4_F32.
// Per (b,oh): out[ow,c] = sum_w Ww[ow,w] * y1[w,c]   (M=7->pad16, K=512, N=64)
// Grid: 112 blocks, 128 threads (4 waves); wave n handles channels [16n,16n+16).
// A fragment (16x4 f32): VGPR0 = {lanes0-15: A[m,K0], lanes16-31: A[m,K2]},
// VGPR1 = K1/K3.  Ww[ow,w] = (w in window) ? 1/sizeW : 0 — synthesized in regs.
// B fragment (4x16 f32): VGPR0 = {lanes0-15: B[K0,n], lanes16-31: B[K2,n]},
// VGPR1 = K1/K3.  C/D 16x16 f32: VGPR r -> M=r (lanes<16) / M=8+r (lanes>=16).
// ---------------------------------------------------------------------------
__global__ __launch_bounds__(128) void pool_w_wmma_kernel(const float* __restrict__ y1,
                                                          float* __restrict__ out) {
    const int boh  = blockIdx.x;                 // b*7 + oh
    const int lane = threadIdx.x & 31;
    const int wave = threadIdx.x >> 5;           // 0..3 -> N tile
    const int n0   = wave * 16;
    const int half = lane >> 4;                  // 0 -> K base +0, 1 -> K base +2
    const int l16  = lane & 15;                  // M (for A) / N (for B) index

    // Per-lane window parameters for A rows (ow = l16; rows 7..15 are zero).
    const int wsL = win_start(l16, W_);
    const int weL = win_stop(l16, W_);
    const float invw = (l16 < OW_) ? 1.0f / (float)(weL - wsL) : 0.0f;

    const float* __restrict__ yb = y1 + (size_t)boh * (W_ * C_);

    v8f c = {};
#pragma unroll 4
    for (int k = 0; k < W_; k += 4) {
        const int kk = k + 2 * half;
        v2f bfrag;
        bfrag.x = yb[(size_t)kk * C_ + n0 + l16];
        bfrag.y = yb[(size_t)(kk + 1) * C_ + n0 + l16];
        v2f afrag;
        afrag.x = (kk     >= wsL && kk     < weL) ? invw : 0.0f;
        afrag.y = (kk + 1 >= wsL && kk + 1 < weL) ? invw : 0.0f;
        c = __builtin_amdgcn_wmma_f32_16x16x4_f32(
                /*neg_a=*/false, afrag, /*neg_b=*/false, bfrag,
                /*c_mod=*/(short)0, c, /*reuse_a=*/false, /*reuse_b=*/false);
    }

    // Write back rows M=0..6 (only lanes 0-15 hold M<8).
    float* __restrict__ ob = out + (size_t)boh * (OW_ * C_);
    if (half == 0) {
#pragma unroll
        for (int r = 0; r < OW_; ++r) {
            ob[(size_t)r * C_ + n0 + l16] = c[r];
        }
    }
}

// ---------------------------------------------------------------------------
// Fallback (only if workspace is too small): direct windowed reduction.
// Grid: B*OH*OW blocks, 256 threads = 16 h-strips x 16 float4-channel groups.
// ---------------------------------------------------------------------------
__global__ __launch_bounds__(256) void pool_direct_kernel(const float* __restrict__ x,
                                                          float* __restrict__ out) {
    __shared__ float4 red[256];
    const int blk = blockIdx.x;
    const int ow  = blk % OW_;
    const int oh  = (blk / OW_) % OH_;
    const int b   = blk / (OH_ * OW_);

    const int c4    = threadIdx.x & 15;   // float4 channel group
    const int strip = threadIdx.x >> 4;   // h strip

    const int h0 = win_start(oh, H_), h1 = win_stop(oh, H_);
    const int w0 = win_start(ow, W_), w1 = win_stop(ow, W_);
    const float inv = 1.0f / ((float)(h1 - h0) * (float)(w1 - w0));

    const float4* __restrict__ xb = (const float4*)x + (size_t)b * (H_ * W_ * (C_ / 4));
    float4 acc = make_float4(0.f, 0.f, 0.f, 0.f);
    for (int h = h0 + strip; h < h1; h += 16) {
        const float4* row = xb + (size_t)h * (W_ * (C_ / 4)) + (size_t)w0 * (C_ / 4) + c4;
        for (int w = w0; w < w1; ++w) {
            float4 v = *row;
            acc.x += v.x; acc.y += v.y; acc.z += v.z; acc.w += v.w;
            row += (C_ / 4);
        }
    }
    red[threadIdx.x] = acc;
    __syncthreads();
    for (int off = 128; off >= 16; off >>= 1) {
        if (threadIdx.x < off) {
            float4 o = red[threadIdx.x + off];
            float4 s = red[threadIdx.x];
            s.x += o.x; s.y += o.y; s.z += o.z; s.w += o.w;
            red[threadIdx.x] = s;
        }
        __syncthreads();
    }
    if (threadIdx.x < 16) {
        float4 s = red[threadIdx.x];
        float4 r = make_float4(s.x * inv, s.y * inv, s.z * inv, s.w * inv);
        float4* ob = (float4*)out + ((size_t)(b * OH_ + oh) * OW_ + ow) * (C_ / 4) + c4;
        *ob = r;
    }
}

extern "C" void kernel_launch(void* const* d_in, const int* in_sizes, int n_in,
                              void* d_out, int out_size, void* d_ws, size_t ws_size,
                              hipStream_t stream) {
    const float* x = (const float*)d_in[0];
    float* out = (float*)d_out;
    (void)in_sizes; (void)n_in; (void)out_size;

    const size_t y1_bytes = (size_t)B_ * OH_ * W_ * C_ * sizeof(float); // 14.68 MB
    if (ws_size >= y1_bytes) {
        float* y1 = (float*)d_ws;
        pool_h_kernel<<<B_ * OH_ * 32, 256, 0, stream>>>(x, y1);
        pool_w_wmma_kernel<<<B_ * OH_, 128, 0, stream>>>(y1, out);
    } else {
        pool_direct_kernel<<<B_ * OH_ * OW_, 256, 0, stream>>>(x, out);
    }
}